// CrossLayerAggregator_79001628443223
// MI455X (gfx1250) — compile-verified
//
#include <hip/hip_runtime.h>

// ---------------------------------------------------------------------------
// Graph attention (2 blocks) for MI455X / gfx1250, wave32.
// GEMMs use V_WMMA_F32_16X16X4_F32 with LDS-staged weights (conflict-free
// padded layout) and a divergence-free inner loop.
// Edge phase (the HBM-roofline bottleneck, ~2.4 GB total) uses wave-per-edge
// float4 gathers + fp32 global atomics for segment sums.
// ---------------------------------------------------------------------------

static constexpr int DIM  = 128;
static constexpr int NBLK = 2;
static constexpr int LDP  = 68;   // padded LDS row stride (floats): banks
                                  // (4*l16 + 2*hk + k) % 64 -> all 32 lanes
                                  // distinct; 16B/8B alignment preserved.

typedef __attribute__((ext_vector_type(2))) float v2f;
typedef __attribute__((ext_vector_type(8))) float v8f;

static __device__ __forceinline__ float wave_reduce_sum(float x) {
#pragma unroll
  for (int off = 16; off > 0; off >>= 1) x += __shfl_xor(x, off, 32);
  return x;
}

// order-preserving float<->uint encoding for atomicMax on floats
static __device__ __forceinline__ unsigned f32_order_enc(float f) {
  unsigned u = __float_as_uint(f);
  return (u & 0x80000000u) ? ~u : (u | 0x80000000u);
}
static __device__ __forceinline__ float f32_order_dec(unsigned e) {
  unsigned u = (e & 0x80000000u) ? (e & 0x7fffffffu) : ~e;
  return __uint_as_float(u);
}

// ---------------------------------------------------------------------------
// C[N x 128] = A[N x 128] @ W^T   with W[128 x 128] row-major.
// Block = 128 threads = 4 waves; each wave owns a 16-row x 128-col strip
// (8 fp32 WMMA accumulators). W is staged through LDS in 128x64 K-chunks.
// A fragment (16x4): lanes 0-15 rows, halves split K (K0,K1 | K2,K3).
// B fragment (4x16): lanes map columns, halves split K; B[k][n] = W[n][k].
// ---------------------------------------------------------------------------
__global__ __launch_bounds__(128) void gemm_nt_128(const float* __restrict__ A,
                                                   const float* __restrict__ W,
                                                   float* __restrict__ C, int N) {
  __shared__ float Wlds[128 * LDP];
  const int tid = threadIdx.x;
  const int wv  = tid >> 5;
  const int ln  = tid & 31;
  const int hk  = ln >> 4;
  const int l16 = ln & 15;
  const int r0  = blockIdx.x * 64 + wv * 16;

  v8f acc[8];
#pragma unroll
  for (int t = 0; t < 8; ++t)
#pragma unroll
    for (int j = 0; j < 8; ++j) acc[t][j] = 0.0f;

  int rowA = r0 + l16;                       // clamp: OOB rows compute garbage,
  if (rowA >= N) rowA = N - 1;               // their stores are guarded below.
  const float* ap = A + (size_t)rowA * DIM;

  for (int kp = 0; kp < DIM; kp += 64) {
    __syncthreads();
    // stage W[:, kp:kp+64] -> LDS (128 rows x 64 cols, padded stride)
#pragma unroll
    for (int i = 0; i < 16; ++i) {
      const int q  = tid + 128 * i;          // float4 index among 2048
      const int n  = q >> 4;                 // 16 float4 per row
      const int c4 = (q & 15) << 2;
      *(float4*)(&Wlds[n * LDP + c4]) =
          *(const float4*)(W + (size_t)n * DIM + kp + c4);
    }
    __syncthreads();

#pragma unroll 4
    for (int k0 = 0; k0 < 64; k0 += 4) {
      const int ka = k0 + 2 * hk;
      v2f a; a.x = ap[kp + ka]; a.y = ap[kp + ka + 1];
#pragma unroll
      for (int t = 0; t < 8; ++t) {
        const float* wp = &Wlds[(t * 16 + l16) * LDP + ka];
        v2f b; b.x = wp[0]; b.y = wp[1];
        acc[t] = __builtin_amdgcn_wmma_f32_16x16x4_f32(
            false, a, false, b, (short)0, acc[t], false, false);
      }
    }
  }

  // C/D layout: vgpr j -> row (r0 + 8*half + j), lane l16 -> column
  const int rb = r0 + hk * 8;
  if (r0 + 16 <= N) {
#pragma unroll
    for (int t = 0; t < 8; ++t)
#pragma unroll
      for (int j = 0; j < 8; ++j)
        C[(size_t)(rb + j) * DIM + t * 16 + l16] = acc[t][j];
  } else {
#pragma unroll
    for (int t = 0; t < 8; ++t)
#pragma unroll
      for (int j = 0; j < 8; ++j) {
        const int row = rb + j;
        if (row < N) C[(size_t)row * DIM + t * 16 + l16] = acc[t][j];
      }
  }
}

// ---------------------------------------------------------------------------
// C[N x 128] = concat(A0[N x 128], A1[N x 128]) @ W^T, W[128 x 256] row-major.
// Same scheme, K = 256 in four LDS-staged 64-wide chunks; A source pointer is
// chunk-uniform (first two chunks from A0, last two from A1).
// ---------------------------------------------------------------------------
__global__ __launch_bounds__(128) void gemm_cat_256(const float* __restrict__ A0,
                                                    const float* __restrict__ A1,
                                                    const float* __restrict__ W,
                                                    float* __restrict__ C, int N) {
  __shared__ float Wlds[128 * LDP];
  const int tid = threadIdx.x;
  const int wv  = tid >> 5;
  const int ln  = tid & 31;
  const int hk  = ln >> 4;
  const int l16 = ln & 15;
  const int r0  = blockIdx.x * 64 + wv * 16;

  v8f acc[8];
#pragma unroll
  for (int t = 0; t < 8; ++t)
#pragma unroll
    for (int j = 0; j < 8; ++j) acc[t][j] = 0.0f;

  int rowA = r0 + l16;
  if (rowA >= N) rowA = N - 1;
  const size_t rbase = (size_t)rowA * DIM;

  for (int kp = 0; kp < 2 * DIM; kp += 64) {
    __syncthreads();
#pragma unroll
    for (int i = 0; i < 16; ++i) {
      const int q  = tid + 128 * i;
      const int n  = q >> 4;
      const int c4 = (q & 15) << 2;
      *(float4*)(&Wlds[n * LDP + c4]) =
          *(const float4*)(W + (size_t)n * (2 * DIM) + kp + c4);
    }
    __syncthreads();

    const float* abase = (kp < DIM) ? (A0 + rbase + kp) : (A1 + rbase + (kp - DIM));
#pragma unroll 4
    for (int k0 = 0; k0 < 64; k0 += 4) {
      const int ka = k0 + 2 * hk;
      v2f a; a.x = abase[ka]; a.y = abase[ka + 1];
#pragma unroll
      for (int t = 0; t < 8; ++t) {
        const float* wp = &Wlds[(t * 16 + l16) * LDP + ka];
        v2f b; b.x = wp[0]; b.y = wp[1];
        acc[t] = __builtin_amdgcn_wmma_f32_16x16x4_f32(
            false, a, false, b, (short)0, acc[t], false, false);
      }
    }
  }

  const int rb = r0 + hk * 8;
  if (r0 + 16 <= N) {
#pragma unroll
    for (int t = 0; t < 8; ++t)
#pragma unroll
      for (int j = 0; j < 8; ++j)
        C[(size_t)(rb + j) * DIM + t * 16 + l16] = acc[t][j];
  } else {
#pragma unroll
    for (int t = 0; t < 8; ++t)
#pragma unroll
      for (int j = 0; j < 8; ++j) {
        const int row = rb + j;
        if (row < N) C[(size_t)row * DIM + t * 16 + l16] = acc[t][j];
      }
  }
}

// ---------------------------------------------------------------------------
// GroupNorm(1, C) over rows: one wave per row (32 lanes x float4 = 128 ch).
// ---------------------------------------------------------------------------
__global__ __launch_bounds__(256) void gn_kernel(const float* __restrict__ X,
                                                 float* __restrict__ Y,
                                                 const float* __restrict__ w,
                                                 const float* __restrict__ b, int N) {
  const int wv = threadIdx.x >> 5, ln = threadIdx.x & 31;
  const int row = blockIdx.x * 8 + wv;
  if (row >= N) return;
  float4 x = ((const float4*)(X + (size_t)row * DIM))[ln];
  float s  = x.x + x.y + x.z + x.w;
  float sq = x.x * x.x + x.y * x.y + x.z * x.z + x.w * x.w;
  s = wave_reduce_sum(s); sq = wave_reduce_sum(sq);
  const float mu = s * (1.0f / DIM);
  const float rs = rsqrtf(sq * (1.0f / DIM) - mu * mu + 1e-5f);
  float4 wv4 = ((const float4*)w)[ln];
  float4 bv4 = ((const float4*)b)[ln];
  float4 y;
  y.x = (x.x - mu) * rs * wv4.x + bv4.x;
  y.y = (x.y - mu) * rs * wv4.y + bv4.y;
  y.z = (x.z - mu) * rs * wv4.z + bv4.z;
  y.w = (x.w - mu) * rs * wv4.w + bv4.w;
  ((float4*)(Y + (size_t)row * DIM))[ln] = y;
}

// GN(gna) -> GN(norm) -> ReLU, fused (one read/write pass)
__global__ __launch_bounds__(256) void gn2_relu_kernel(const float* __restrict__ X,
                                                       float* __restrict__ Y,
                                                       const float* __restrict__ w1,
                                                       const float* __restrict__ b1,
                                                       const float* __restrict__ w2,
                                                       const float* __restrict__ b2,
                                                       int N) {
  const int wv = threadIdx.x >> 5, ln = threadIdx.x & 31;
  const int row = blockIdx.x * 8 + wv;
  if (row >= N) return;
  float4 x = ((const float4*)(X + (size_t)row * DIM))[ln];
  float s  = x.x + x.y + x.z + x.w;
  float sq = x.x * x.x + x.y * x.y + x.z * x.z + x.w * x.w;
  s = wave_reduce_sum(s); sq = wave_reduce_sum(sq);
  float mu = s * (1.0f / DIM);
  float rs = rsqrtf(sq * (1.0f / DIM) - mu * mu + 1e-5f);
  float4 wa = ((const float4*)w1)[ln], ba = ((const float4*)b1)[ln];
  float4 y;
  y.x = (x.x - mu) * rs * wa.x + ba.x;
  y.y = (x.y - mu) * rs * wa.y + ba.y;
  y.z = (x.z - mu) * rs * wa.z + ba.z;
  y.w = (x.w - mu) * rs * wa.w + ba.w;
  // second GN
  s  = y.x + y.y + y.z + y.w;
  sq = y.x * y.x + y.y * y.y + y.z * y.z + y.w * y.w;
  s = wave_reduce_sum(s); sq = wave_reduce_sum(sq);
  mu = s * (1.0f / DIM);
  rs = rsqrtf(sq * (1.0f / DIM) - mu * mu + 1e-5f);
  float4 wn = ((const float4*)w2)[ln], bn = ((const float4*)b2)[ln];
  float4 z;
  z.x = fmaxf((y.x - mu) * rs * wn.x + bn.x, 0.0f);
  z.y = fmaxf((y.y - mu) * rs * wn.y + bn.y, 0.0f);
  z.z = fmaxf((y.z - mu) * rs * wn.z + bn.z, 0.0f);
  z.w = fmaxf((y.w - mu) * rs * wn.w + bn.w, 0.0f);
  ((float4*)(Y + (size_t)row * DIM))[ln] = z;
}

// out = relu(GN(y) + res), fused epilogue of the Wlin GEMM
__global__ __launch_bounds__(256) void gn_res_relu_kernel(const float* __restrict__ X,
                                                          const float* __restrict__ R,
                                                          float* __restrict__ Y,
                                                          const float* __restrict__ w,
                                                          const float* __restrict__ b,
                                                          int N) {
  const int wv = threadIdx.x >> 5, ln = threadIdx.x & 31;
  const int row = blockIdx.x * 8 + wv;
  if (row >= N) return;
  float4 x = ((const float4*)(X + (size_t)row * DIM))[ln];
  float s  = x.x + x.y + x.z + x.w;
  float sq = x.x * x.x + x.y * x.y + x.z * x.z + x.w * x.w;
  s = wave_reduce_sum(s); sq = wave_reduce_sum(sq);
  const float mu = s * (1.0f / DIM);
  const float rs = rsqrtf(sq * (1.0f / DIM) - mu * mu + 1e-5f);
  float4 wv4 = ((const float4*)w)[ln], bv4 = ((const float4*)b)[ln];
  float4 r = ((const float4*)(R + (size_t)row * DIM))[ln];
  float4 y;
  y.x = fmaxf((x.x - mu) * rs * wv4.x + bv4.x + r.x, 0.0f);
  y.y = fmaxf((x.y - mu) * rs * wv4.y + bv4.y + r.y, 0.0f);
  y.z = fmaxf((x.z - mu) * rs * wv4.z + bv4.z + r.z, 0.0f);
  y.w = fmaxf((x.w - mu) * rs * wv4.w + bv4.w + r.w, 0.0f);
  ((float4*)(Y + (size_t)row * DIM))[ln] = y;
}

// ---------------------------------------------------------------------------
// Edge phase. Wave-per-edge: 32 lanes x float4 == one 512B row gather.
// ---------------------------------------------------------------------------
__global__ __launch_bounds__(256) void edge_dot_kernel(const float* __restrict__ Q,
                                                       const float* __restrict__ K,
                                                       const int* __restrict__ u,
                                                       const int* __restrict__ v,
                                                       float* __restrict__ c,
                                                       unsigned* __restrict__ gmax,
                                                       int E) {
  __shared__ float smax[8];
  const int wv = threadIdx.x >> 5, ln = threadIdx.x & 31;
  const int e = blockIdx.x * 8 + wv;
  float cv = -3.0e38f;
  if (e < E) {
    const int vi = v[e], ui = u[e];
    float4 qv = ((const float4*)(Q + (size_t)vi * DIM))[ln];
    float4 kv = ((const float4*)(K + (size_t)ui * DIM))[ln];
    float d = qv.x * kv.x + qv.y * kv.y + qv.z * kv.z + qv.w * kv.w;
    d = wave_reduce_sum(d);
    d = (d >= 0.0f) ? d : 0.1f * d;      // LeakyReLU(0.1)
    if (ln == 0) c[e] = d;
    cv = d;
  }
  if (ln == 0) smax[wv] = cv;
  __syncthreads();
  if (threadIdx.x == 0) {
    float m = smax[0];
#pragma unroll
    for (int i = 1; i < 8; ++i) m = fmaxf(m, smax[i]);
    atomicMax(gmax, f32_order_enc(m));
  }
}

__global__ __launch_bounds__(256) void edge_exp_kernel(float* __restrict__ c,
                                                       float* __restrict__ csum,
                                                       const int* __restrict__ v,
                                                       const unsigned* __restrict__ gmax,
                                                       int E) {
  const int e = blockIdx.x * 256 + threadIdx.x;
  if (e >= E) return;
  const float m = f32_order_dec(*gmax);
  const float ce = expf(c[e] - m);
  c[e] = ce;
  atomicAdd(&csum[v[e]], ce);
}

__global__ __launch_bounds__(256) void edge_scatter_kernel(const float* __restrict__ c,
                                                           const float* __restrict__ csum,
                                                           const float* __restrict__ ctxp,
                                                           const int* __restrict__ u,
                                                           const int* __restrict__ v,
                                                           float* __restrict__ agg,
                                                           int E) {
  const int wv = threadIdx.x >> 5, ln = threadIdx.x & 31;
  const int e = blockIdx.x * 8 + wv;
  if (e >= E) return;
  const int vi = v[e], ui = u[e];
  const float coef = c[e] / (csum[vi] + 1e-16f);
  float4 m = ((const float4*)(ctxp + (size_t)ui * DIM))[ln];
  float* dst = agg + (size_t)vi * DIM + ln * 4;
  atomicAdd(dst + 0, m.x * coef);
  atomicAdd(dst + 1, m.y * coef);
  atomicAdd(dst + 2, m.z * coef);
  atomicAdd(dst + 3, m.w * coef);
}

__global__ void zero_f32(float* __restrict__ p, long n) {
  const long i = (long)blockIdx.x * 256 + threadIdx.x;
  if (i < n) p[i] = 0.0f;
}
__global__ void init_scalar_u32(unsigned* p) {
  if (threadIdx.x == 0) *p = 0u;   // encodes "most negative float"
}

// ---------------------------------------------------------------------------
extern "C" void kernel_launch(void* const* d_in, const int* in_sizes, int n_in,
                              void* d_out, int out_size, void* d_ws, size_t ws_size,
                              hipStream_t stream) {
  (void)in_sizes; (void)n_in; (void)out_size; (void)ws_size;
  const int NA = 50000, NC = 100000, NE = 600000;

  const float* agt    = (const float*)d_in[0];
  const float* ctx    = (const float*)d_in[1];
  const int*   u      = (const int*)d_in[2];
  const int*   v      = (const int*)d_in[3];
  const float* Wq     = (const float*)d_in[4];
  const float* Wk     = (const float*)d_in[5];
  const float* Wctx   = (const float*)d_in[6];
  const float* gnc_w  = (const float*)d_in[7];
  const float* gnc_b  = (const float*)d_in[8];
  const float* Wagt   = (const float*)d_in[9];
  const float* gna_w  = (const float*)d_in[10];
  const float* gna_b  = (const float*)d_in[11];
  const float* norm_w = (const float*)d_in[12];
  const float* norm_b = (const float*)d_in[13];
  const float* Wlin   = (const float*)d_in[14];
  const float* gnl_w  = (const float*)d_in[15];
  const float* gnl_b  = (const float*)d_in[16];

  float* ws = (float*)d_ws;
  size_t off = 0;
  float* qbuf   = ws + off; off += (size_t)NA * DIM;   // q; reused for y=x@Wlin^T
  float* kbuf   = ws + off; off += (size_t)NC * DIM;   // k
  float* cpbuf  = ws + off; off += (size_t)NC * DIM;   // ctx_p
  float* cbuf   = ws + off; off += (size_t)NE;         // edge coeffs
  float* csum   = ws + off; off += (size_t)NA;         // segment sums
  float* aggbuf = ws + off; off += (size_t)NA * DIM;   // aggregated messages
  float* xbuf   = ws + off; off += (size_t)NA * DIM;   // hidden
  float* agtbuf = ws + off; off += (size_t)NA * DIM;   // agt ping-pong
  unsigned* maxbuf = (unsigned*)(ws + off); off += 16;

  const dim3 b128(128), b256(256);

  for (int i = 0; i < NBLK; ++i) {
    const float* ain  = (i == 0) ? agt : agtbuf;
    float*       aout = (i == NBLK - 1) ? (float*)d_out : agtbuf;

    // projections (WMMA fp32)
    gemm_nt_128<<<(NA + 63) / 64, b128, 0, stream>>>(ain, Wq + (size_t)i * DIM * DIM, qbuf, NA);
    gemm_nt_128<<<(NC + 63) / 64, b128, 0, stream>>>(ctx, Wk + (size_t)i * DIM * DIM, kbuf, NC);
    gemm_nt_128<<<(NC + 63) / 64, b128, 0, stream>>>(ctx, Wctx + (size_t)i * DIM * DIM, cpbuf, NC);
    gn_kernel<<<(NC + 7) / 8, b256, 0, stream>>>(cpbuf, cpbuf, gnc_w + i * DIM, gnc_b + i * DIM, NC);

    // edge attention
    init_scalar_u32<<<1, 32, 0, stream>>>(maxbuf);
    zero_f32<<<(NA + 255) / 256, b256, 0, stream>>>(csum, (long)NA);
    zero_f32<<<((long)NA * DIM + 255) / 256, b256, 0, stream>>>(aggbuf, (long)NA * DIM);
    edge_dot_kernel<<<(NE + 7) / 8, b256, 0, stream>>>(qbuf, kbuf, u, v, cbuf, maxbuf, NE);
    edge_exp_kernel<<<(NE + 255) / 256, b256, 0, stream>>>(cbuf, csum, v, maxbuf, NE);
    edge_scatter_kernel<<<(NE + 7) / 8, b256, 0, stream>>>(cbuf, csum, cpbuf, u, v, aggbuf, NE);

    // update MLP
    gemm_cat_256<<<(NA + 63) / 64, b128, 0, stream>>>(ain, aggbuf, Wagt + (size_t)i * DIM * 2 * DIM, xbuf, NA);
    gn2_relu_kernel<<<(NA + 7) / 8, b256, 0, stream>>>(xbuf, xbuf, gna_w + i * DIM, gna_b + i * DIM,
                                                       norm_w + i * DIM, norm_b + i * DIM, NA);
    gemm_nt_128<<<(NA + 63) / 64, b128, 0, stream>>>(xbuf, Wlin + (size_t)i * DIM * DIM, qbuf, NA);
    gn_res_relu_kernel<<<(NA + 7) / 8, b256, 0, stream>>>(qbuf, ain, aout, gnl_w + i * DIM, gnl_b + i * DIM, NA);
  }
}